// Sampling_Module_82970178224270
// MI455X (gfx1250) — compile-verified
//
#include <hip/hip_runtime.h>
#include <hip/hip_bf16.h>
#include <math.h>

// ---------------------------------------------------------------------------
// Problem constants (from the reference)
// ---------------------------------------------------------------------------
#define BATCH   2
#define N_KEY   64
#define N_CAND  64
#define NPTS    (N_KEY * N_CAND)   // 4096
#define N_TGT   8192
#define C_DEEP  32
#define KNN     32
#define OUT_C   35                 // 3 xyz + 32 feat
#define RADIUS2 1.0f

#define CHUNK   512                // targets per LDS score tile
#define NCHUNK  (N_TGT / CHUNK)    // 16
#define NTILES  (N_TGT / 16)       // 512 WMMA B tiles total
#define TPC     (CHUNK / 16)       // 32 tiles per chunk
#define EPC     (TPC * 32)         // 1024 v2f elements per chunk (8 KB)

#define FLT_BIG 3.0e38f

typedef __attribute__((ext_vector_type(2))) float v2f;
typedef __attribute__((ext_vector_type(8))) float v8f;

// ---------------------------------------------------------------------------
// CDNA5 async helpers: ASYNCcnt-tracked global->LDS copy + drain.
// ---------------------------------------------------------------------------
__device__ __forceinline__ void async_g2lds_b64(unsigned ldsOff, const void* gptr) {
    asm volatile("global_load_async_to_lds_b64 %0, %1, off"
                 :: "v"(ldsOff), "v"(gptr) : "memory");
}

__device__ __forceinline__ void wait_async0() {
#if __has_builtin(__builtin_amdgcn_s_wait_asynccnt)
    __builtin_amdgcn_s_wait_asynccnt(0);
#else
    asm volatile("s_wait_asynccnt 0x0" ::: "memory");
#endif
}

// ---------------------------------------------------------------------------
// Kernel 0: pack the WMMA B' operand once (batch-independent).
// B' = (x, y, z, |b|^2) laid out exactly as the 4x16 f32 B operand expects:
//   element e = tile*32 + lane ; lane<16 -> (x,y) of target tile*16+lane,
//                                lane>=16 -> (z,|b|^2) of target tile*16+lane-16.
// ---------------------------------------------------------------------------
__global__ __launch_bounds__(256) void pack_b_kernel(const float* __restrict__ tgtXyz,
                                                     float2* __restrict__ bop) {
    const int e = blockIdx.x * 256 + threadIdx.x;
    if (e >= NTILES * 32) return;
    const int tile = e >> 5;
    const int lane = e & 31;
    const int tgt  = tile * 16 + (lane & 15);
    const float x = tgtXyz[tgt * 3 + 0];
    const float y = tgtXyz[tgt * 3 + 1];
    const float z = tgtXyz[tgt * 3 + 2];
    float2 v;
    if (lane < 16) { v.x = x; v.y = y; }
    else           { v.x = z; v.y = x * x + y * y + z * z; }
    bop[e] = v;
}

// ---------------------------------------------------------------------------
// Kernel 1: farthest point sampling (full permutation, npoint == N).
// One workgroup per batch; sequential scan with LDS argmax reduction.
// ---------------------------------------------------------------------------
__global__ __launch_bounds__(256) void fps_kernel(const float* __restrict__ cand,
                                                  int* __restrict__ fpsIdx,
                                                  float* __restrict__ newXyz) {
    __shared__ float dist[NPTS];
    __shared__ float rv[256];
    __shared__ int   ri[256];
    __shared__ int   sFar;

    const int b   = blockIdx.x;
    const int tid = threadIdx.x;
    const float* xyz = cand + (size_t)b * NPTS * 3;

    for (int i = tid; i < NPTS; i += 256) dist[i] = 1.0e10f;
    if (tid == 0) sFar = 0;
    __syncthreads();

    for (int s = 0; s < NPTS; ++s) {
        const int far = sFar;
        const float cx = xyz[far * 3 + 0];
        const float cy = xyz[far * 3 + 1];
        const float cz = xyz[far * 3 + 2];
        if (tid == 0) {
            fpsIdx[b * NPTS + s] = far;
            newXyz[((size_t)b * NPTS + s) * 3 + 0] = cx;
            newXyz[((size_t)b * NPTS + s) * 3 + 1] = cy;
            newXyz[((size_t)b * NPTS + s) * 3 + 2] = cz;
        }
        float bv = -1.0f; int bi = 0;
        for (int i = tid; i < NPTS; i += 256) {
            const float dx = xyz[i * 3 + 0] - cx;
            const float dy = xyz[i * 3 + 1] - cy;
            const float dz = xyz[i * 3 + 2] - cz;
            const float d  = dx * dx + dy * dy + dz * dz;
            const float nd = fminf(dist[i], d);
            dist[i] = nd;
            if (nd > bv) { bv = nd; bi = i; }   // strict > keeps lowest index
        }
        rv[tid] = bv; ri[tid] = bi;
        __syncthreads();
        for (int off = 128; off > 0; off >>= 1) {
            if (tid < off) {
                const float va = rv[tid], vb = rv[tid + off];
                const int   ia = ri[tid], ib = ri[tid + off];
                if (vb > va || (vb == va && ib < ia)) { rv[tid] = vb; ri[tid] = ib; }
            }
            __syncthreads();
        }
        if (tid == 0) sFar = ri[0];
        __syncthreads();
    }
}

// ---------------------------------------------------------------------------
// Kernel 2: radius grouping -> out[..., 0:3].
// K smallest indices with sqdist <= R^2, padded with the first hit.
// ---------------------------------------------------------------------------
__global__ __launch_bounds__(256) void group_kernel(const float* __restrict__ cand,
                                                    const float* __restrict__ newXyz,
                                                    float* __restrict__ out) {
    const int g = blockIdx.x * 256 + threadIdx.x;      // b*NPTS + s
    if (g >= BATCH * NPTS) return;
    const int b = g / NPTS;
    const float* xyz = cand + (size_t)b * NPTS * 3;

    const float qx = newXyz[(size_t)g * 3 + 0];
    const float qy = newXyz[(size_t)g * 3 + 1];
    const float qz = newXyz[(size_t)g * 3 + 2];

    int idxs[KNN];
    int cnt = 0;
    for (int j = 0; j < NPTS && cnt < KNN; ++j) {
        const float dx = xyz[j * 3 + 0] - qx;
        const float dy = xyz[j * 3 + 1] - qy;
        const float dz = xyz[j * 3 + 2] - qz;
        if (dx * dx + dy * dy + dz * dz <= RADIUS2) idxs[cnt++] = j;
    }
    if (cnt == 0) { idxs[0] = 0; cnt = 1; }
    const int first = idxs[0];
    for (int k = cnt; k < KNN; ++k) idxs[k] = first;

    float* o = out + (size_t)g * KNN * OUT_C;
    for (int k = 0; k < KNN; ++k) {
        const int j = idxs[k];
        o[k * OUT_C + 0] = xyz[j * 3 + 0] - qx;
        o[k * OUT_C + 1] = xyz[j * 3 + 1] - qy;
        o[k * OUT_C + 2] = xyz[j * 3 + 2] - qz;
    }
}

// ---------------------------------------------------------------------------
// Kernel 3: kNN + weighted feature gather -> out[..., 3:35].
// WMMA score = -2*a.b + |b|^2 (order-preserving per row). B' operand slabs are
// double-buffered through LDS with GLOBAL_LOAD_ASYNC_TO_LDS_B64 / s_wait_asynccnt.
// Top-K: the running best-32 list lives in REGISTERS (one entry per lane,
// maintained with shuffles); each argmin pass finds the lexicographic
// (score, index) successor of the previous pick, so the LDS score tile is
// read-only after the barrier — pure ds_load traffic, no volatile/flat.
// ---------------------------------------------------------------------------
__global__ __launch_bounds__(256) void knn_feat_kernel(const float* __restrict__ cand,
                                                       const float2* __restrict__ bop,
                                                       const float* __restrict__ tgtFeat,
                                                       float* __restrict__ out) {
    __shared__ float sc[16 * CHUNK];       // 32 KB score tile
    __shared__ v2f   bopLds[2 * EPC];      // 16 KB double-buffered B' slabs
    __shared__ float fDist[16][KNN];       // final per-row neighbor distances
    __shared__ int   fIdx[16][KNN];        // final per-row neighbor indices
    __shared__ float aaS[16];
    __shared__ float wsum[16];

    const int tid  = threadIdx.x;
    const int lane = tid & 31;
    const int wv   = tid >> 5;             // 8 waves
    const int blk  = blockIdx.x;           // BATCH * (NPTS/16)
    const int b    = blk >> 8;
    const int rowBase = (blk & 255) * 16;

    const float* xyz = cand + (size_t)b * NPTS * 3;
    const unsigned bopLdsBase = (unsigned)(uintptr_t)&bopLds[0];

    if (tid < 16) {
        const int r = rowBase + tid;
        const float x = xyz[r * 3 + 0], y = xyz[r * 3 + 1], z = xyz[r * 3 + 2];
        aaS[tid] = x * x + y * y + z * z;
    }

    // A operand (16x4 f32, 2 VGPRs): lanes 0-15 -> K=0,1 ; lanes 16-31 -> K=2,3
    const int mr = rowBase + (lane & 15);
    const float ax = xyz[mr * 3 + 0], ay = xyz[mr * 3 + 1], az = xyz[mr * 3 + 2];
    v2f A;
    if (lane < 16) { A.x = -2.0f * ax; A.y = -2.0f * ay; }
    else           { A.x = -2.0f * az; A.y = 1.0f;       }

    // register-resident best lists: lane k == list entry k, rows 2*wv and 2*wv+1
    float bV[2] = { FLT_BIG, FLT_BIG };
    int   bI[2] = { 0, 0 };

    // prologue: async-stage chunk 0's B' slab into buffer 0
#pragma unroll
    for (int i = 0; i < EPC / 256; ++i) {
        const int e = i * 256 + tid;
        async_g2lds_b64(bopLdsBase + (unsigned)e * 8u, bop + e);
    }

    for (int ch = 0; ch < NCHUNK; ++ch) {
        const int chBase = ch * CHUNK;
        const int buf    = ch & 1;

        wait_async0();                     // slab for this chunk landed
        __syncthreads();

        // prefetch next chunk's slab into the other buffer (overlaps selection)
        if (ch + 1 < NCHUNK) {
#pragma unroll
            for (int i = 0; i < EPC / 256; ++i) {
                const int e = i * 256 + tid;
                async_g2lds_b64(bopLdsBase + (unsigned)((buf ^ 1) * EPC + e) * 8u,
                                bop + (size_t)(ch + 1) * EPC + e);
            }
        }

        // ---- score tile: each wave computes 4 of the 32 target tiles
        for (int t = wv; t < TPC; t += 8) {
            const v2f Bm = bopLds[buf * EPC + t * 32 + lane];

            v8f acc = {};
            acc = __builtin_amdgcn_wmma_f32_16x16x4_f32(
                /*neg_a=*/false, A, /*neg_b=*/false, Bm,
                /*c_mod=*/(short)0, acc, /*reuse_a=*/false, /*reuse_b=*/false);

            const int colBase = t * 16 + (lane & 15);
            const int rOff    = (lane < 16) ? 0 : 8;
#pragma unroll
            for (int r = 0; r < 8; ++r)
                sc[(r + rOff) * CHUNK + colBase] = acc[r];
        }
        __syncthreads();

        // ---- wave-local top-K over the (read-only) score tile
#pragma unroll
        for (int rr = 0; rr < 2; ++rr) {
            const int r = wv * 2 + rr;
            float lastV = -FLT_BIG;
            int   lastJ = -1;
            for (int p = 0; p < KNN; ++p) {
                // lexicographic successor of (lastV, lastJ) in this chunk
                float mv = FLT_BIG; int mj = CHUNK;
                for (int j = lane; j < CHUNK; j += 32) {
                    const float v = sc[r * CHUNK + j];
                    const bool after = (v > lastV) || (v == lastV && j > lastJ);
                    if (after && (v < mv || (v == mv && j < mj))) { mv = v; mj = j; }
                }
#pragma unroll
                for (int off = 16; off > 0; off >>= 1) {
                    const float ov = __shfl_xor(mv, off, 32);
                    const int   oj = __shfl_xor(mj, off, 32);
                    if (ov < mv || (ov == mv && oj < mj)) { mv = ov; mj = oj; }
                }
                const int gIdx = chBase + mj;

                // stop if it cannot displace the current 32nd-best (lane 31)
                const float tV = __shfl(bV[rr], 31, 32);
                const int   tI = __shfl(bI[rr], 31, 32);
                if (mv > tV || (mv == tV && gIdx > tI)) break;

                // parallel sorted insert into the register list via shuffles
                float pV = __shfl_up(bV[rr], 1, 32);
                int   pI = __shfl_up(bI[rr], 1, 32);
                if (lane == 0) { pV = -FLT_BIG; pI = -1; }
                const bool keep  = (bV[rr] < mv) || (bV[rr] == mv && bI[rr] < gIdx);
                const bool pKeep = (pV     < mv) || (pV     == mv && pI     < gIdx);
                bV[rr] = keep ? bV[rr] : (pKeep ? mv   : pV);
                bI[rr] = keep ? bI[rr] : (pKeep ? gIdx : pI);

                lastV = mv; lastJ = mj;
            }
        }
        __syncthreads();
    }

    // ---- score -> true distance, weight denominator, publish lists to LDS
#pragma unroll
    for (int rr = 0; rr < 2; ++rr) {
        const int r = wv * 2 + rr;
        const float d = sqrtf(fmaxf(bV[rr] + aaS[r], 0.0f));
        float sum = d;
#pragma unroll
        for (int off = 16; off > 0; off >>= 1) sum += __shfl_xor(sum, off, 32);
        fDist[r][lane] = d;
        fIdx[r][lane]  = bI[rr];
        if (lane == 0) wsum[r] = sum;
    }
    __syncthreads();

    // ---- gather + scale features: lane == channel, one (row,k) pair per wave/iter
    for (int p = wv; p < 16 * KNN; p += 8) {
        const int r = p >> 5;
        const int k = p & 31;
        const int s = rowBase + r;
        const float w  = fDist[r][k] / wsum[r];
        const int   ti = fIdx[r][k];
        const float f  = tgtFeat[((size_t)b * N_TGT + ti) * C_DEEP + lane];
        out[(((size_t)b * NPTS + s) * KNN + k) * OUT_C + 3 + lane] = f * w;
    }
}

// ---------------------------------------------------------------------------
// Launch
// ---------------------------------------------------------------------------
extern "C" void kernel_launch(void* const* d_in, const int* in_sizes, int n_in,
                              void* d_out, int out_size, void* d_ws, size_t ws_size,
                              hipStream_t stream) {
    const float* cand    = (const float*)d_in[0];   // (B,64,64,3)
    // d_in[1] = src_keypts: unused by the reference computation
    const float* tgtXyz  = (const float*)d_in[2];   // (1,8192,3)
    const float* tgtFeat = (const float*)d_in[3];   // (B,8192,32)
    float* out = (float*)d_out;                     // (B,64,64,32,35)

    // workspace layout: fps indices | gathered queries | packed B' operand
    int*    fpsIdx = (int*)d_ws;
    float*  newXyz = (float*)((char*)d_ws + (size_t)BATCH * NPTS * sizeof(int));
    float2* bop    = (float2*)((char*)d_ws + (size_t)BATCH * NPTS * sizeof(int)
                                           + (size_t)BATCH * NPTS * 3 * sizeof(float));

    pack_b_kernel<<<(NTILES * 32 + 255) / 256, 256, 0, stream>>>(tgtXyz, bop);
    fps_kernel<<<BATCH, 256, 0, stream>>>(cand, fpsIdx, newXyz);
    group_kernel<<<(BATCH * NPTS + 255) / 256, 256, 0, stream>>>(cand, newXyz, out);
    knn_feat_kernel<<<BATCH * (NPTS / 16), 256, 0, stream>>>(cand, (const float2*)bop,
                                                             tgtFeat, out);
}